// ScaledDotProductAttention_2619930050597
// MI455X (gfx1250) — compile-verified
//
#include <hip/hip_runtime.h>
#include <math.h>

#define BB   8
#define SS   2048
#define DD   768
#define HH1  512
#define NCH  16            // column-stats chunks along q
#define CHQ  (SS / NCH)    // 128 rows per chunk

typedef __attribute__((ext_vector_type(2))) float v2f;
typedef __attribute__((ext_vector_type(8))) float v8f;

__device__ __forceinline__ v2f ld2(const float* p) {
    return *reinterpret_cast<const v2f*>(p);
}

// ---------------------------------------------------------------------------
// K1: gate g[row] = tanh(v_row @ W1 + b1) @ W2 + b2
// fc1 via f32 WMMA: block = 16 rows, 8 waves x (16x64 h-strip), K = 768.
// Hidden activations parked in LDS; fc2 = deterministic tree reduction.
// ---------------------------------------------------------------------------
__global__ __launch_bounds__(256) void gate_kernel(const float* __restrict__ v,
                                                   const float* __restrict__ W1,
                                                   const float* __restrict__ b1,
                                                   const float* __restrict__ W2,
                                                   const float* __restrict__ b2,
                                                   float* __restrict__ g) {
    __shared__ float hid[16 * HH1];   // 32 KB
    __shared__ float red[16 * 256];   // 16 KB
    const int tid  = threadIdx.x;
    const int wave = tid >> 5;
    const int lane = tid & 31;
    const int lm   = lane & 15;
    const int koff = (lane >> 4) * 2;     // A: lanes 0-15 K=0,1; 16-31 K=2,3
    const long row0 = (long)blockIdx.x * 16;

    const float* ap = v + (row0 + lm) * DD + koff;
    const int h0 = wave * 64 + lm;        // this wave's h columns: h0 + t*16

    v8f c[4] = {{}, {}, {}, {}};
#pragma unroll 2
    for (int kk = 0; kk < DD; kk += 4) {
        v2f a = ld2(ap + kk);
        const float* wb = W1 + (long)(kk + koff) * HH1 + h0;
#pragma unroll
        for (int t = 0; t < 4; ++t) {
            v2f bt;
            bt.x = wb[t * 16];
            bt.y = wb[HH1 + t * 16];
            c[t] = __builtin_amdgcn_wmma_f32_16x16x4_f32(false, a, false, bt,
                                                         (short)0, c[t], false, false);
        }
    }
    const int rowoff = (lane >> 4) << 3;  // C: lanes 16-31 hold M = r + 8
#pragma unroll
    for (int t = 0; t < 4; ++t) {
        const int h = h0 + t * 16;
        const float bias = b1[h];
#pragma unroll
        for (int r = 0; r < 8; ++r)
            hid[(rowoff + r) * HH1 + h] = tanhf(c[t][r] + bias);
    }
    __syncthreads();

    // fc2: per-row dot with W2 over 512 h
    float p[16];
#pragma unroll
    for (int r = 0; r < 16; ++r) p[r] = 0.f;
#pragma unroll
    for (int hh = 0; hh < HH1 / 256; ++hh) {
        const int h = tid + hh * 256;
        const float w2 = W2[h];
#pragma unroll
        for (int r = 0; r < 16; ++r) p[r] += hid[r * HH1 + h] * w2;
    }
#pragma unroll
    for (int r = 0; r < 16; ++r) red[r * 256 + tid] = p[r];
    __syncthreads();
    for (int st = 128; st > 0; st >>= 1) {
        if (tid < st)
#pragma unroll
            for (int r = 0; r < 16; ++r)
                red[r * 256 + tid] += red[r * 256 + tid + st];
        __syncthreads();
    }
    if (tid < 16) g[row0 + tid] = red[tid * 256] + b2[0];
}

// ---------------------------------------------------------------------------
// K2: scores = q @ k^T / sqrt(S) into the attn_weights region of d_out.
// One wave per 32x32 tile: 4 accumulators, 2 A + 2 B fragments per K-step
// (1.0 global b64 load per WMMA; operands are L2-resident on MI455X).
// ---------------------------------------------------------------------------
__global__ __launch_bounds__(32) void scores_kernel(const float* __restrict__ q,
                                                    const float* __restrict__ k,
                                                    float* __restrict__ W) {
    const int lane  = threadIdx.x;
    const int b     = blockIdx.z;
    const int qbase = blockIdx.y * 32;
    const int kbase = blockIdx.x * 32;
    const int lm    = lane & 15;
    const int koff  = (lane >> 4) * 2;

    const float* qp0 = q + ((long)(b * SS + qbase + lm)) * DD + koff;
    const float* qp1 = qp0 + 16 * DD;
    const float* kp0 = k + ((long)(b * SS + kbase + lm)) * DD + koff;
    const float* kp1 = kp0 + 16 * DD;

    v8f c00 = {}, c01 = {}, c10 = {}, c11 = {};
#pragma unroll 4
    for (int kk = 0; kk < DD; kk += 4) {
        v2f a0 = ld2(qp0 + kk);
        v2f a1 = ld2(qp1 + kk);
        v2f b0 = ld2(kp0 + kk);
        v2f b1v = ld2(kp1 + kk);
        c00 = __builtin_amdgcn_wmma_f32_16x16x4_f32(false, a0, false, b0,
                                                    (short)0, c00, false, false);
        c01 = __builtin_amdgcn_wmma_f32_16x16x4_f32(false, a0, false, b1v,
                                                    (short)0, c01, false, false);
        c10 = __builtin_amdgcn_wmma_f32_16x16x4_f32(false, a1, false, b0,
                                                    (short)0, c10, false, false);
        c11 = __builtin_amdgcn_wmma_f32_16x16x4_f32(false, a1, false, b1v,
                                                    (short)0, c11, false, false);
    }

    const float scale = 0.022097086912079608f;   // 1/sqrt(2048)
    const int row0 = qbase + ((lane >> 4) << 3);
#pragma unroll
    for (int r = 0; r < 8; ++r) {
        long i0 = ((long)(b * SS + row0 + r)) * SS + kbase + lm;
        long i1 = i0 + (long)16 * SS;
        W[i0]      = c00[r] * scale;
        W[i0 + 16] = c01[r] * scale;
        W[i1]      = c10[r] * scale;
        W[i1 + 16] = c11[r] * scale;
    }
}

// ---------------------------------------------------------------------------
// K3a: chunked online column softmax stats (axis=1). 16x more parallelism
// than one-thread-per-column; strided walk helped by global_prefetch.
// ---------------------------------------------------------------------------
__global__ __launch_bounds__(256) void colstats_part(const float* __restrict__ W,
                                                     float* __restrict__ pm,
                                                     float* __restrict__ pl) {
    const int b    = blockIdx.y;
    const int c    = blockIdx.z;
    const int kcol = blockIdx.x * 256 + threadIdx.x;
    const float* base = W + (long)b * SS * SS + (long)c * CHQ * SS + kcol;
    float m = -3.4e38f, l = 0.f;
    for (int qq = 0; qq < CHQ; ++qq) {
        const int pq = (qq + 6 < CHQ) ? qq + 6 : CHQ - 1;
        __builtin_prefetch(base + (long)pq * SS, 0, 0);
        const float x  = base[(long)qq * SS];
        const float mn = fmaxf(m, x);
        l = l * __expf(m - mn) + __expf(x - mn);
        m = mn;
    }
    pm[((long)b * NCH + c) * SS + kcol] = m;
    pl[((long)b * NCH + c) * SS + kcol] = l;
}

// K3b: combine the NCH partial (m, l) pairs per column (exact).
__global__ __launch_bounds__(256) void colstats_combine(const float* __restrict__ pm,
                                                        const float* __restrict__ pl,
                                                        float* __restrict__ colM,
                                                        float* __restrict__ colL) {
    const int b    = blockIdx.y;
    const int kcol = blockIdx.x * 256 + threadIdx.x;
    float m = -3.4e38f;
#pragma unroll
    for (int c = 0; c < NCH; ++c)
        m = fmaxf(m, pm[((long)b * NCH + c) * SS + kcol]);
    float l = 0.f;
#pragma unroll
    for (int c = 0; c < NCH; ++c)
        l += pl[((long)b * NCH + c) * SS + kcol] *
             __expf(pm[((long)b * NCH + c) * SS + kcol] - m);
    colM[b * SS + kcol] = m;
    colL[b * SS + kcol] = l;
}

// ---------------------------------------------------------------------------
// K4: per-row transform, in place:
//   s  = exp(score - colM)/colL            (softmax over axis=1)
//   t  = (s*(1-g) + g/S) / sqrt(768)       (s_mean == 1/S analytically)
//   row= softmax_k(t)
// ---------------------------------------------------------------------------
__global__ __launch_bounds__(256) void rowsoftmax_kernel(float* __restrict__ W,
                                                         const float* __restrict__ g,
                                                         const float* __restrict__ colM,
                                                         const float* __restrict__ colL) {
    __shared__ float t[SS];
    __shared__ float red[256];
    const int tid = threadIdx.x;
    const long rid = blockIdx.x;            // flattened b*S + q
    const int b = (int)(rid / SS);
    float* row = W + rid * SS;

    const float invSq768 = 0.036084391824351615f;  // 1/sqrt(768)
    const float invS     = 1.0f / SS;

    float lmax = -3.4e38f;
#pragma unroll
    for (int i = 0; i < SS / 256; ++i) {
        const int kc = tid + i * 256;
        const float x  = row[kc];
        const float gv = g[b * SS + kc];
        const float sc = __expf(x - colM[b * SS + kc]) / colL[b * SS + kc];
        const float tt = (sc * (1.f - gv) + gv * invS) * invSq768;
        t[kc] = tt;
        lmax = fmaxf(lmax, tt);
    }
    red[tid] = lmax; __syncthreads();
    for (int st = 128; st > 0; st >>= 1) {
        if (tid < st) red[tid] = fmaxf(red[tid], red[tid + st]);
        __syncthreads();
    }
    const float rmax = red[0];
    __syncthreads();

    float e[SS / 256]; float lsum = 0.f;
#pragma unroll
    for (int i = 0; i < SS / 256; ++i) {
        e[i] = __expf(t[tid + i * 256] - rmax);
        lsum += e[i];
    }
    red[tid] = lsum; __syncthreads();
    for (int st = 128; st > 0; st >>= 1) {
        if (tid < st) red[tid] += red[tid + st];
        __syncthreads();
    }
    const float inv = 1.f / red[0];
#pragma unroll
    for (int i = 0; i < SS / 256; ++i) row[tid + i * 256] = e[i] * inv;
}

// ---------------------------------------------------------------------------
// K5: attn_output = attn_weights @ v + q.  32x32 tile per wave, K = 2048.
// ---------------------------------------------------------------------------
__global__ __launch_bounds__(32) void outgemm_kernel(const float* __restrict__ W,
                                                     const float* __restrict__ v,
                                                     const float* __restrict__ qin,
                                                     float* __restrict__ out) {
    const int lane  = threadIdx.x;
    const int b     = blockIdx.z;
    const int qbase = blockIdx.y * 32;
    const int dbase = blockIdx.x * 32;
    const int lm    = lane & 15;
    const int koff  = (lane >> 4) * 2;

    const float* ap0 = W + ((long)(b * SS + qbase + lm)) * SS + koff;
    const float* ap1 = ap0 + (long)16 * SS;
    const float* vp  = v + ((long)b * SS) * DD + dbase + lm;

    v8f c00 = {}, c01 = {}, c10 = {}, c11 = {};
#pragma unroll 4
    for (int kk = 0; kk < SS; kk += 4) {
        v2f a0 = ld2(ap0 + kk);
        v2f a1 = ld2(ap1 + kk);
        const float* vb = vp + (long)(kk + koff) * DD;
        v2f b0, b1v;
        b0.x  = vb[0];       b0.y  = vb[DD];
        b1v.x = vb[16];      b1v.y = vb[DD + 16];
        c00 = __builtin_amdgcn_wmma_f32_16x16x4_f32(false, a0, false, b0,
                                                    (short)0, c00, false, false);
        c01 = __builtin_amdgcn_wmma_f32_16x16x4_f32(false, a0, false, b1v,
                                                    (short)0, c01, false, false);
        c10 = __builtin_amdgcn_wmma_f32_16x16x4_f32(false, a1, false, b0,
                                                    (short)0, c10, false, false);
        c11 = __builtin_amdgcn_wmma_f32_16x16x4_f32(false, a1, false, b1v,
                                                    (short)0, c11, false, false);
    }

    const int row0 = qbase + ((lane >> 4) << 3);
#pragma unroll
    for (int r = 0; r < 8; ++r) {
        long i0 = ((long)(b * SS + row0 + r)) * DD + dbase + lm;
        long i1 = i0 + (long)16 * DD;
        out[i0]      = c00[r] + qin[i0];
        out[i0 + 16] = c01[r] + qin[i0 + 16];
        out[i1]      = c10[r] + qin[i1];
        out[i1 + 16] = c11[r] + qin[i1 + 16];
    }
}

// ---------------------------------------------------------------------------
extern "C" void kernel_launch(void* const* d_in, const int* in_sizes, int n_in,
                              void* d_out, int out_size, void* d_ws, size_t ws_size,
                              hipStream_t stream) {
    (void)in_sizes; (void)n_in; (void)out_size; (void)ws_size;
    const float* q  = (const float*)d_in[0];
    const float* k  = (const float*)d_in[1];
    const float* v  = (const float*)d_in[2];
    const float* W1 = (const float*)d_in[3];
    const float* b1 = (const float*)d_in[4];
    const float* W2 = (const float*)d_in[5];
    const float* b2 = (const float*)d_in[6];

    float* out      = (float*)d_out;
    float* attn_out = out;                              // [B,S,D]
    float* Wt       = out + (long)BB * SS * DD;         // [B,S,S] scores->weights
    float* g    = (float*)d_ws;                         // B*S
    float* colM = g + BB * SS;                          // B*S
    float* colL = colM + BB * SS;                       // B*S
    float* pm   = colL + BB * SS;                       // B*NCH*S
    float* pl   = pm + (long)BB * NCH * SS;             // B*NCH*S

    gate_kernel<<<dim3(BB * SS / 16), dim3(256), 0, stream>>>(v, W1, b1, W2, b2, g);
    scores_kernel<<<dim3(SS / 32, SS / 32, BB), dim3(32), 0, stream>>>(q, k, Wt);
    colstats_part<<<dim3(SS / 256, BB, NCH), dim3(256), 0, stream>>>(Wt, pm, pl);
    colstats_combine<<<dim3(SS / 256, BB), dim3(256), 0, stream>>>(pm, pl, colM, colL);
    rowsoftmax_kernel<<<dim3(BB * SS), dim3(256), 0, stream>>>(Wt, g, colM, colL);
    outgemm_kernel<<<dim3(DD / 32, SS / 32, BB), dim3(32), 0, stream>>>(Wt, v, q, attn_out);
}